// PerfusionMapLoss_62852551409806
// MI455X (gfx1250) — compile-verified
//
#include <hip/hip_runtime.h>

typedef __attribute__((ext_vector_type(2))) float v2f;
typedef __attribute__((ext_vector_type(8))) float v8f;

#define HW     65536
#define TDIM   60
#define NVOL   8
#define NBASE  6

__device__ __forceinline__ const float* vol_base(const float* pred, const float* tgt, int v) {
    return ((v < 4) ? pred : tgt) + (size_t)(v & 3) * TDIM * HW;
}

// CDNA5 async copy: global -> LDS, 16B per active lane, tracked by ASYNCcnt.
__device__ __forceinline__ void async_load_b128(unsigned lds_off, const void* gaddr) {
    asm volatile("global_load_async_to_lds_b128 %0, %1, off"
                 :: "v"(lds_off), "v"(gaddr) : "memory");
}

// s_wait_asynccnt with an exact immediate (async loads complete in order).
__device__ __forceinline__ void wait_async_le(int n) {
    switch (n) {
        case 0: asm volatile("s_wait_asynccnt 0x0" ::: "memory"); break;
        case 1: asm volatile("s_wait_asynccnt 0x1" ::: "memory"); break;
        case 2: asm volatile("s_wait_asynccnt 0x2" ::: "memory"); break;
        case 3: asm volatile("s_wait_asynccnt 0x3" ::: "memory"); break;
        case 4: asm volatile("s_wait_asynccnt 0x4" ::: "memory"); break;
        case 5: asm volatile("s_wait_asynccnt 0x5" ::: "memory"); break;
        case 6: asm volatile("s_wait_asynccnt 0x6" ::: "memory"); break;
        case 7: asm volatile("s_wait_asynccnt 0x7" ::: "memory"); break;
        default: asm volatile("s_wait_asynccnt 0x8" ::: "memory"); break;
    }
}

// ---------------------------------------------------------------------------
// Kernel 1: per-pixel temporal stats (max, argmax, baseline, cbv).
// float4-vectorized streaming: grid (64, 8) x 256 threads.
// ---------------------------------------------------------------------------
__global__ __launch_bounds__(256) void stats_kernel(
    const float* __restrict__ pred, const float* __restrict__ tgt,
    const float* __restrict__ tv,
    float* __restrict__ maps, float* __restrict__ maxval,
    float* __restrict__ arrival, float* __restrict__ baseline) {
    const int v = blockIdx.y;
    const int p4 = blockIdx.x * blockDim.x + threadIdx.x;     // float4 index
    const float4* vol4 = (const float4*)vol_base(pred, tgt, v);

    float mx[4] = {-1e30f, -1e30f, -1e30f, -1e30f};
    int   am[4] = {0, 0, 0, 0};
    float bs[4] = {0.f, 0.f, 0.f, 0.f};
    for (int t = 0; t < TDIM; ++t) {
        float4 x = vol4[(size_t)t * (HW / 4) + p4];
        float xs[4] = {x.x, x.y, x.z, x.w};
#pragma unroll
        for (int j = 0; j < 4; ++j) {
            if (xs[j] > mx[j]) { mx[j] = xs[j]; am[j] = t; }
            if (t < NBASE) bs[j] += xs[j];
        }
    }
#pragma unroll
    for (int j = 0; j < 4; ++j) bs[j] *= (1.f / NBASE);

    const float dt = tv[1] - tv[0];
    float cbv[4] = {0.f, 0.f, 0.f, 0.f};
    for (int t = 0; t < TDIM; ++t) {
        float4 x = vol4[(size_t)t * (HW / 4) + p4];
        float xs[4] = {x.x, x.y, x.z, x.w};
#pragma unroll
        for (int j = 0; j < 4; ++j) {
            float d = xs[j] - bs[j];
            cbv[j] += (d > 0.f) ? d : 0.f;
        }
    }

    float4* mv4 = (float4*)(maxval + (size_t)v * HW);
    float4* ar4 = (float4*)(arrival + (size_t)v * HW);
    float4* bl4 = (float4*)(baseline + (size_t)v * HW);
    float4* cb4 = (float4*)(maps + ((size_t)(v * 4) + 1) * HW);
    mv4[p4] = make_float4(mx[0], mx[1], mx[2], mx[3]);
    ar4[p4] = make_float4((float)am[0], (float)am[1], (float)am[2], (float)am[3]);
    bl4[p4] = make_float4(bs[0], bs[1], bs[2], bs[3]);
    cb4[p4] = make_float4(cbv[0] * dt, cbv[1] * dt, cbv[2] * dt, cbv[3] * dt);
}

// ---------------------------------------------------------------------------
// Kernel 2: 0.95-quantile of maxval via exact radix select on float bits
// (values >= 0 -> uint bit pattern is order-preserving). One block/volume.
// ---------------------------------------------------------------------------
__global__ __launch_bounds__(256) void quantile_kernel(
    const float* __restrict__ maxval, float* __restrict__ thr) {
    const int v = blockIdx.x;
    const float* mv = maxval + (size_t)v * HW;
    __shared__ unsigned hist[256];
    __shared__ unsigned s_byte, s_rem;

    const double pos = 0.95 * (double)(HW - 1);   // 62258.25
    const unsigned lo = (unsigned)pos;
    const float frac = (float)(pos - (double)lo);

    float vals[2];
    for (int k = 0; k < 2; ++k) {
        unsigned prefix = 0u;
        unsigned rem = lo + (unsigned)k;
        for (int shift = 24; shift >= 0; shift -= 8) {
            hist[threadIdx.x] = 0u;
            __syncthreads();
            unsigned maskHigh = (shift == 24) ? 0u : ((~0u) << (shift + 8));
            for (int i = threadIdx.x; i < HW; i += blockDim.x) {
                unsigned u = __float_as_uint(mv[i]);
                if ((u & maskHigh) == prefix) atomicAdd(&hist[(u >> shift) & 0xFF], 1u);
            }
            __syncthreads();
            if (threadIdx.x == 0) {
                unsigned run = 0;
                for (unsigned b = 0; b < 256; ++b) {
                    unsigned h = hist[b];
                    if (run + h > rem) { s_byte = b; s_rem = rem - run; break; }
                    run += h;
                }
            }
            __syncthreads();
            prefix |= (s_byte << shift);
            rem = s_rem;
            __syncthreads();
        }
        vals[k] = __uint_as_float(prefix);
    }
    if (threadIdx.x == 0) thr[v] = vals[0] + frac * (vals[1] - vals[0]);
}

// ---------------------------------------------------------------------------
// Kernel 3: masked AIF sums per time-point (+ mask count once), float4 loads.
// grid (60, 8) x 256 threads. Mask arrays are L2-resident.
// ---------------------------------------------------------------------------
__global__ __launch_bounds__(256) void aif_sum_kernel(
    const float* __restrict__ pred, const float* __restrict__ tgt,
    const float* __restrict__ maxval, const float* __restrict__ arrival,
    const float* __restrict__ thr,
    float* __restrict__ aifsum, float* __restrict__ cntv) {
    const int t = blockIdx.x, v = blockIdx.y, tid = threadIdx.x;
    const float4* vol4 = (const float4*)(vol_base(pred, tgt, v) + (size_t)t * HW);
    const float4* mv4 = (const float4*)(maxval + (size_t)v * HW);
    const float4* ar4 = (const float4*)(arrival + (size_t)v * HW);
    const float th = thr[v];
    __shared__ float rs[256], rc[256];

    float s = 0.f, c = 0.f;
    for (int i = tid; i < HW / 4; i += 256) {
        float4 a = ar4[i];
        float4 m = mv4[i];
        float4 x = vol4[i];
        float as[4] = {a.x, a.y, a.z, a.w};
        float ms[4] = {m.x, m.y, m.z, m.w};
        float xs[4] = {x.x, x.y, x.z, x.w};
#pragma unroll
        for (int j = 0; j < 4; ++j) {
            if (as[j] < (float)TDIM * 0.3f && ms[j] > th) { s += xs[j]; c += 1.f; }
        }
    }
    rs[tid] = s; rc[tid] = c;
    __syncthreads();
    for (int o = 128; o > 0; o >>= 1) {
        if (tid < o) { rs[tid] += rs[tid + o]; rc[tid] += rc[tid + o]; }
        __syncthreads();
    }
    if (tid == 0) {
        aifsum[v * 64 + t] = rs[0];
        if (t == 0) cntv[v] = rc[0];
    }
}

// ---------------------------------------------------------------------------
// Kernel 4: finalize AIF (masked vs center fallback, baseline relu, area norm).
// ---------------------------------------------------------------------------
__global__ __launch_bounds__(64) void aif_finalize_kernel(
    const float* __restrict__ pred, const float* __restrict__ tgt,
    const float* __restrict__ tv,
    const float* __restrict__ aifsum, const float* __restrict__ cntv,
    float* __restrict__ aif) {
    const int v = blockIdx.x, t = threadIdx.x;
    const float* vol = vol_base(pred, tgt, v);
    __shared__ float a[64];
    __shared__ float red[64];

    const float cnt = cntv[v];
    float val = 0.f;
    if (t < TDIM) {
        if (cnt > 0.f) {
            val = aifsum[v * 64 + t] / fmaxf(cnt, 1.f);
        } else {
            float s = 0.f;
            for (int r = 126; r < 130; ++r)
                for (int c = 126; c < 130; ++c)
                    s += vol[(size_t)t * HW + r * 256 + c];
            val = s * (1.f / 16.f);
        }
    }
    a[t] = val;
    __syncthreads();
    const float bs = (a[0] + a[1] + a[2] + a[3] + a[4] + a[5]) * (1.f / NBASE);
    val = (t < TDIM) ? fmaxf(val - bs, 0.f) : 0.f;
    red[t] = val;
    __syncthreads();
    for (int s = 32; s > 0; s >>= 1) {
        if (t < s) red[t] += red[t + s];
        __syncthreads();
    }
    const float dt = tv[1] - tv[0];
    const float area = red[0] * dt;
    if (area > 0.f) val /= area;
    aif[v * 64 + t] = val;
}

// ---------------------------------------------------------------------------
// Kernel 5: 60x60 SVD-thresholded pseudo-inverse via one-sided Jacobi.
// SINGLE WAVE per volume: per-wave LDS ordering -> zero barriers; dot
// products via butterfly __shfl_xor. Output padded to 64x64 for WMMA.
// ---------------------------------------------------------------------------
__global__ __launch_bounds__(32) void svd_pinv_kernel(
    const float* __restrict__ aif, float* __restrict__ invA) {
    const int v = blockIdx.x, tid = threadIdx.x;   // 0..31, single wave
    __shared__ float Wm[3600];
    __shared__ float Vm[3600];
    __shared__ float aifs[64];
    __shared__ float sigS[64];
    __shared__ float invs2[64];

    aifs[tid]      = aif[v * 64 + tid];
    aifs[tid + 32] = aif[v * 64 + tid + 32];
    for (int idx = tid; idx < 3600; idx += 32) {
        int r = idx % TDIM, c = idx / TDIM;
        Wm[idx] = (c <= r) ? aifs[c] : 0.f;       // A[r][c] = aif[c] * tril
        Vm[idx] = (r == c) ? 1.f : 0.f;
    }

    for (int sweep = 0; sweep < 10; ++sweep) {
        for (int p = 0; p < TDIM - 1; ++p) {
            for (int q = p + 1; q < TDIM; ++q) {
                const int r0 = tid, r1 = tid + 32;
                float wp0 = Wm[r0 + p * TDIM], wq0 = Wm[r0 + q * TDIM];
                float wp1 = (r1 < TDIM) ? Wm[r1 + p * TDIM] : 0.f;
                float wq1 = (r1 < TDIM) ? Wm[r1 + q * TDIM] : 0.f;
                float al = wp0 * wp0 + wp1 * wp1;
                float be = wq0 * wq0 + wq1 * wq1;
                float ga = wp0 * wq0 + wp1 * wq1;
                for (int o = 16; o > 0; o >>= 1) {
                    al += __shfl_xor(al, o);
                    be += __shfl_xor(be, o);
                    ga += __shfl_xor(ga, o);
                }
                if (fabsf(ga) > 1e-12f * sqrtf(al * be) + 1e-30f) {  // wave-uniform
                    float zeta = (be - al) / (2.f * ga);
                    float tt = ((zeta >= 0.f) ? 1.f : -1.f) /
                               (fabsf(zeta) + sqrtf(1.f + zeta * zeta));
                    float cs = rsqrtf(1.f + tt * tt);
                    float sn = cs * tt;
                    Wm[r0 + p * TDIM] = cs * wp0 - sn * wq0;
                    Wm[r0 + q * TDIM] = sn * wp0 + cs * wq0;
                    float vp0 = Vm[r0 + p * TDIM], vq0 = Vm[r0 + q * TDIM];
                    Vm[r0 + p * TDIM] = cs * vp0 - sn * vq0;
                    Vm[r0 + q * TDIM] = sn * vp0 + cs * vq0;
                    if (r1 < TDIM) {
                        Wm[r1 + p * TDIM] = cs * wp1 - sn * wq1;
                        Wm[r1 + q * TDIM] = sn * wp1 + cs * wq1;
                        float vp1 = Vm[r1 + p * TDIM], vq1 = Vm[r1 + q * TDIM];
                        Vm[r1 + p * TDIM] = cs * vp1 - sn * vq1;
                        Vm[r1 + q * TDIM] = sn * vp1 + cs * vq1;
                    }
                }
            }
        }
    }

    for (int j = tid; j < 64; j += 32) {
        float s = 0.f;
        if (j < TDIM)
            for (int r = 0; r < TDIM; ++r) { float w = Wm[r + j * TDIM]; s += w * w; }
        sigS[j] = sqrtf(s);
    }
    float m = fmaxf(sigS[tid], sigS[tid + 32]);
    for (int o = 16; o > 0; o >>= 1) m = fmaxf(m, __shfl_xor(m, o));
    const float thresh = 0.2f * m;                 // SVD_REG
    for (int j = tid; j < 64; j += 32) {
        float s = sigS[j];
        invs2[j] = (j < TDIM && s > thresh) ? 1.f / (s * s) : 0.f;
    }
    for (int idx = tid; idx < 4096; idx += 32) {
        int i = idx >> 6, k = idx & 63;
        float s = 0.f;
        if (i < TDIM && k < TDIM)
            for (int j = 0; j < TDIM; ++j)
                s += Vm[i + j * TDIM] * Wm[k + j * TDIM] * invs2[j];
        invA[(size_t)v * 4096 + idx] = s;
    }
}

// ---------------------------------------------------------------------------
// Kernel 6: residue = relu(invA @ relu(vol - baseline)) fused with per-pixel
// max/argmax -> cbf, tmax.  V_WMMA_F32_16X16X4_F32 with ASYNC-to-LDS staging:
//   - invA (64x64) staged block-wide with 4 async B128 loads / thread group,
//   - each wave stages its own 64x16 B-tile with 8 async B128 chunk loads
//     (8 rows x 16 px each), all 12 async ops in flight simultaneously,
//   - chunk c consumed after s_wait_asynccnt <= 7-c (in-order completion).
// Rows 60..63 are zero-padded in LDS; chunk 7's OOB lanes are EXEC-masked.
// ---------------------------------------------------------------------------
__global__ __launch_bounds__(256) void gemm_maps_kernel(
    const float* __restrict__ pred, const float* __restrict__ tgt,
    const float* __restrict__ tv,
    const float* __restrict__ invA, const float* __restrict__ baseline,
    float* __restrict__ maps) {
    const int v = blockIdx.y;
    const float* vol = vol_base(pred, tgt, v);
    __shared__ float As[64 * 64];        // invA, row-major [m][k]   (16 KB)
    __shared__ float Bt[8][64 * 16];     // per-wave B tile [row][px] (32 KB)

    const int lane = threadIdx.x & 31;
    const int wave = threadIdx.x >> 5;
    const int colbase = blockIdx.x * 128 + wave * 16;
    const int col  = colbase + (lane & 15);
    const int hi   = lane >> 4;
    const int koff = hi * 2;

    // ---- stage invA asynchronously (4 x B128 per thread-quad of the block)
    {
        const int base = threadIdx.x * 4;                 // float index, 16B aligned
#pragma unroll
        for (int it = 0; it < 4; ++it) {
            async_load_b128((unsigned)(uintptr_t)&As[base + it * 1024],
                            invA + (size_t)v * 4096 + base + it * 1024);
        }
    }

    // ---- zero-pad B rows 60..63, then stage rows 0..59 asynchronously
    Bt[wave][960 + lane] = 0.f;                           // rows 60..63
    Bt[wave][992 + lane] = 0.f;
    const int lrow = lane >> 2;                           // 0..7
    const int lpx  = (lane & 3) * 4;                      // 0,4,8,12
    const unsigned ldsOff0 =
        (unsigned)(uintptr_t)&Bt[wave][0] + (unsigned)(lrow * 16 + lpx) * 4u;
    const float* gbase = vol + (size_t)lrow * HW + colbase + lpx;
#pragma unroll
    for (int c = 0; c < 7; ++c) {                         // rows 0..55
        async_load_b128(ldsOff0 + (unsigned)(c * 8 * 16 * 4),
                        gbase + (size_t)(c * 8) * HW);
    }
    if (lrow < 4) {                                       // rows 56..59 only
        async_load_b128(ldsOff0 + (unsigned)(7 * 8 * 16 * 4),
                        gbase + (size_t)56 * HW);
    }

    wait_async_le(8);            // the 4 invA loads (issued first) are done
    __syncthreads();             // As visible block-wide

    const float bl = baseline[(size_t)v * HW + col];

    v8f acc[4] = {};
#pragma unroll
    for (int c = 0; c < 8; ++c) {
        wait_async_le(7 - c);    // B chunk c (rows 8c..8c+7) has landed
#pragma unroll
        for (int kc = 0; kc < 2; ++kc) {
            const int kk = c * 2 + kc;
            const int r0 = kk * 4 + koff;
            v2f b;
            b.x = fmaxf(Bt[wave][r0 * 16 + (lane & 15)]       - bl, 0.f);
            b.y = fmaxf(Bt[wave][(r0 + 1) * 16 + (lane & 15)] - bl, 0.f);
            if (r0     >= TDIM) b.x = 0.f;
            if (r0 + 1 >= TDIM) b.y = 0.f;
#pragma unroll
            for (int mt = 0; mt < 4; ++mt) {
                v2f a;
                const int arow = mt * 16 + (lane & 15);
                a.x = As[arow * 64 + kk * 4 + koff];
                a.y = As[arow * 64 + kk * 4 + koff + 1];
                acc[mt] = __builtin_amdgcn_wmma_f32_16x16x4_f32(
                    false, a, false, b, (short)0, acc[mt], false, false);
            }
        }
    }

    // per-lane max/argmax over the rows this lane holds (ascending order ->
    // strict '>' keeps first occurrence like jnp.argmax)
    float best = -1.f;
    int bidx = 0;
#pragma unroll
    for (int mt = 0; mt < 4; ++mt) {
#pragma unroll
        for (int vv = 0; vv < 8; ++vv) {
            const int row = mt * 16 + vv + hi * 8;
            if (row < TDIM) {
                float val = fmaxf(acc[mt][vv], 0.f);      // relu(residue)
                if (val > best) { best = val; bidx = row; }
            }
        }
    }
    const float obest = __shfl_down(best, 16);
    const int   oidx  = __shfl_down(bidx, 16);
    if (lane < 16) {
        if (obest > best || (obest == best && oidx < bidx)) { best = obest; bidx = oidx; }
        maps[((size_t)(v * 4) + 0) * HW + col] = best;        // cbf
        maps[((size_t)(v * 4) + 3) * HW + col] = tv[bidx];    // tmax
    }
}

// ---------------------------------------------------------------------------
// Loss kernels
// ---------------------------------------------------------------------------
__global__ void zero_kernel(float* __restrict__ out, int n) {
    int i = blockIdx.x * blockDim.x + threadIdx.x;
    if (i < n) out[i] = 0.f;
}

__global__ __launch_bounds__(256) void loss_maps_kernel(
    const float* __restrict__ maps, float* __restrict__ out) {
    __shared__ float red[256];
    const int tid = threadIdx.x;
    float acc = 0.f;
    for (int idx = blockIdx.x * blockDim.x + tid; idx < 4 * HW;
         idx += gridDim.x * blockDim.x) {
        const int b = idx >> 16;
        const int p = idx & (HW - 1);
        const size_t vp = (size_t)b * 4, vt = (size_t)(b + 4) * 4;
        const float cbfp = maps[(vp + 0) * HW + p], cbft = maps[(vt + 0) * HW + p];
        const float cbvp = maps[(vp + 1) * HW + p], cbvt = maps[(vt + 1) * HW + p];
        const float tmxp = maps[(vp + 3) * HW + p], tmxt = maps[(vt + 3) * HW + p];
        const float mttp = (cbfp > 0.f) ? cbvp / cbfp : 0.f;
        const float mttt = (cbft > 0.f) ? cbvt / cbft : 0.f;
        acc += 1.0f * fabsf(cbfp - cbft) + 1.0f * fabsf(tmxp - tmxt) +
               0.7f * fabsf(cbvp - cbvt) + 0.5f * fabsf(mttp - mttt);
    }
    red[tid] = acc * (1.f / (4.f * HW));
    __syncthreads();
    for (int s = 128; s > 0; s >>= 1) {
        if (tid < s) red[tid] += red[tid + s];
        __syncthreads();
    }
    if (tid == 0) atomicAdd(out, red[0]);
}

__global__ __launch_bounds__(256) void loss_mse_kernel(
    const float* __restrict__ pred, const float* __restrict__ tgt,
    float* __restrict__ out) {
    __shared__ float red[256];
    const int tid = threadIdx.x;
    const size_t N4 = (size_t)TDIM * HW;   // float4 count = 4*T*HW/4
    const float4* p4 = (const float4*)pred;
    const float4* t4 = (const float4*)tgt;
    float acc = 0.f;
    for (size_t idx = (size_t)blockIdx.x * blockDim.x + tid; idx < N4;
         idx += (size_t)gridDim.x * blockDim.x) {
        float4 a = p4[idx], b = t4[idx];
        float d0 = a.x - b.x, d1 = a.y - b.y, d2 = a.z - b.z, d3 = a.w - b.w;
        acc += d0 * d0 + d1 * d1 + d2 * d2 + d3 * d3;
    }
    red[tid] = acc * (0.3f / (float)((size_t)4 * TDIM * HW));
    __syncthreads();
    for (int s = 128; s > 0; s >>= 1) {
        if (tid < s) red[tid] += red[tid + s];
        __syncthreads();
    }
    if (tid == 0) atomicAdd(out, red[0]);
}

// ---------------------------------------------------------------------------
// Launcher
// ---------------------------------------------------------------------------
extern "C" void kernel_launch(void* const* d_in, const int* in_sizes, int n_in,
                              void* d_out, int out_size, void* d_ws, size_t ws_size,
                              hipStream_t stream) {
    const float* pred = (const float*)d_in[0];
    const float* tgt  = (const float*)d_in[1];
    const float* tv   = (const float*)d_in[2];
    float* out = (float*)d_out;

    float* ws       = (float*)d_ws;
    float* maps     = ws;                               // [8][4][HW]
    float* maxval   = maps + (size_t)NVOL * 4 * HW;     // [8][HW]
    float* arrival  = maxval + (size_t)NVOL * HW;       // [8][HW]
    float* baseline = arrival + (size_t)NVOL * HW;      // [8][HW]
    float* invA     = baseline + (size_t)NVOL * HW;     // [8][64*64]
    float* thr      = invA + (size_t)NVOL * 4096;       // [8]
    float* cntv     = thr + NVOL;                       // [8]
    float* aifsum   = cntv + NVOL;                      // [8][64]
    float* aif      = aifsum + NVOL * 64;               // [8][64]

    stats_kernel<<<dim3(HW / 4 / 256, NVOL), 256, 0, stream>>>(
        pred, tgt, tv, maps, maxval, arrival, baseline);
    quantile_kernel<<<NVOL, 256, 0, stream>>>(maxval, thr);
    aif_sum_kernel<<<dim3(TDIM, NVOL), 256, 0, stream>>>(
        pred, tgt, maxval, arrival, thr, aifsum, cntv);
    aif_finalize_kernel<<<NVOL, 64, 0, stream>>>(pred, tgt, tv, aifsum, cntv, aif);
    svd_pinv_kernel<<<NVOL, 32, 0, stream>>>(aif, invA);
    gemm_maps_kernel<<<dim3(HW / 128, NVOL), 256, 0, stream>>>(
        pred, tgt, tv, invA, baseline, maps);
    zero_kernel<<<1, 64, 0, stream>>>(out, out_size);
    loss_maps_kernel<<<128, 256, 0, stream>>>(maps, out);
    loss_mse_kernel<<<512, 256, 0, stream>>>(pred, tgt, out);
}